// PhotonicNetworkGPU_51857435131955
// MI455X (gfx1250) — compile-verified
//
#include <hip/hip_runtime.h>
#include <math.h>

typedef __attribute__((ext_vector_type(2))) float v2f;
typedef __attribute__((ext_vector_type(4))) float v4f;
typedef __attribute__((ext_vector_type(8))) float v8f;

// Problem constants
#define S_DIM 256
#define W_DIM 64
#define M_DIM 64
#define B_DIM 32
#define NCOL 16           // columns per workgroup
#define STEPS 4
#define IMAG_OFF 33554432 // B*S*W*M

// LDS layout: (row-pair p, col c) -> 4 contiguous floats:
// [ re(2p,c), re(2p+1,c), im(2p,c), im(2p+1,c) ]
#define PADDR(p, c) (((((p) << 4) + (c)) << 2))

__global__ __launch_bounds__(512)
void photonic_net_kernel(const float* __restrict__ field_real,
                         const float* __restrict__ field_imag,
                         const float* __restrict__ conn,
                         const float* __restrict__ gain,
                         const float* __restrict__ bias,
                         const float* __restrict__ ww,     // (S, W)
                         const float* __restrict__ disp,   // (W,)
                         float* __restrict__ out)          // (2, B, S, W, M)
{
    __shared__ float lf[8192];   // 128 pairs x 16 cols x 4 floats = 32 KB

    const int tid  = threadIdx.x;
    const int lane = tid & 31;
    const int wid  = tid >> 5;     // wave id = output row-tile (0..15)
    const int h    = lane >> 4;    // lane half
    const int cl   = lane & 15;    // col in tile / row in A-fragment

    const int ncol0 = blockIdx.x * NCOL;    // global column index
    const int b     = ncol0 >> 12;          // / (W*M)
    const int nb    = ncol0 & 4095;         // column within batch
    const int w     = nb >> 6;              // wavelength index (same for all 16 cols)

    // ---- stage field columns into LDS (pair-interleaved re/im) ----
    {
        const int c  = tid & 15;
        const int p0 = tid >> 4;            // 32 pairs per pass, 128 pairs total
        #pragma unroll
        for (int it = 0; it < 4; ++it) {
            int p = p0 + it * 32;
            int s0 = 2 * p;
            size_t g0 = ((size_t)(b * S_DIM + s0) << 12) + (size_t)(nb + c);
            v4f v;
            v.x = field_real[g0];
            v.y = field_real[g0 + 4096];
            v.z = field_imag[g0];
            v.w = field_imag[g0 + 4096];
            *(v4f*)(&lf[PADDR(p, c)]) = v;
        }
    }

    // ---- load A = (conn + I) row-slab into registers, WMMA 16x4 f32 layout ----
    // lane<16: holds (i=16*wid+cl, k0), (i, k0+1); lane>=16: (i, k0+2), (i, k0+3)
    const int i = wid * 16 + cl;
    v2f a[64];
    #pragma unroll
    for (int t = 0; t < 64; ++t) {
        int k = 4 * t + 2 * h;
        v2f av = *(const v2f*)(conn + (size_t)i * S_DIM + k);
        av.x += (i == k)     ? 1.0f : 0.0f;   // fold field + coupled into (I+C)
        av.y += (i == k + 1) ? 1.0f : 0.0f;
        // Force materialization of the folded value: without this the compiler
        // rematerializes the +I fold (cmp/cndmask/add) at every use, every step.
        asm volatile("" : "+v"(av));
        a[t] = av;
    }

    // ---- per-row (per-vgpr) constants: gain, bias, decay; per-w complex phase ----
    float g8[8], b8[8], d8[8];
    #pragma unroll
    for (int v = 0; v < 8; ++v) {
        int s = wid * 16 + v + 8 * h;       // D-matrix row held in vgpr v
        g8[v] = gain[s];
        b8[v] = bias[s];
        d8[v] = 0.95f * ww[s * W_DIM + w];  // (1 - CAVITY_LOSS) * wavelength_weights
    }
    const float ang = 0.1f * disp[w];       // DISPERSION_STRENGTH * dispersion_shape
    const float phr = cosf(ang);
    const float phi = sinf(ang);

    __syncthreads();

    for (int step = 0; step < STEPS; ++step) {
        v8f ar = {0.f,0.f,0.f,0.f,0.f,0.f,0.f,0.f};
        v8f ai = {0.f,0.f,0.f,0.f,0.f,0.f,0.f,0.f};

        // coupled' = (I + conn) @ field over K=256, 64 x (K=4) WMMA steps.
        // 4-deep explicit prefetch pipeline: one ds_load_b128 per k-block,
        // issued 4 blocks ahead so waits run at dscnt<=3.
        v4f fb[4];
        #pragma unroll
        for (int t = 0; t < 4; ++t)
            fb[t] = *(const v4f*)(&lf[PADDR(2 * t + h, cl)]);

        #pragma unroll
        for (int t = 0; t < 64; ++t) {
            v4f f = fb[t & 3];
            if (t + 4 < 64)
                fb[t & 3] = *(const v4f*)(&lf[PADDR(2 * (t + 4) + h, cl)]);
            v2f br = __builtin_shufflevector(f, f, 0, 1);
            v2f bi = __builtin_shufflevector(f, f, 2, 3);
            ar = __builtin_amdgcn_wmma_f32_16x16x4_f32(
                     false, a[t], false, br, (short)0, ar, false, false);
            ai = __builtin_amdgcn_wmma_f32_16x16x4_f32(
                     false, a[t], false, bi, (short)0, ai, false, false);
        }

        __syncthreads();   // everyone done READING old field before anyone writes

        #pragma unroll
        for (int v = 0; v < 8; ++v) {
            // field = g*(field+coupled) + b   (b is real: adds to real part only)
            float fr = g8[v] * ar[v] + b8[v];
            float fi = g8[v] * ai[v];
            // saturation: denominator >= 1, v_rcp_f32 (~1 ulp) is safe here
            float inten = fr * fr + fi * fi;
            float sc = __builtin_amdgcn_rcpf(1.0f + inten * 0.5f);
            fr *= sc;
            fi *= sc;
            // complex phase * real decay
            float nr = (fr * phr - fi * phi) * d8[v];
            float ni = (fr * phi + fi * phr) * d8[v];

            int s = wid * 16 + v + 8 * h;
            if (step == STEPS - 1) {
                size_t g = ((size_t)(b * S_DIM + s) << 12) + (size_t)(nb + cl);
                out[g]            = nr;
                out[g + IMAG_OFF] = ni;
            } else {
                int p = s >> 1, par = s & 1;
                lf[PADDR(p, cl) + par]     = nr;
                lf[PADDR(p, cl) + 2 + par] = ni;
            }
        }

        __syncthreads();   // writes visible before next step's reads
    }
}

extern "C" void kernel_launch(void* const* d_in, const int* in_sizes, int n_in,
                              void* d_out, int out_size, void* d_ws, size_t ws_size,
                              hipStream_t stream) {
    const float* field_real = (const float*)d_in[0];
    const float* field_imag = (const float*)d_in[1];
    const float* conn       = (const float*)d_in[2];
    const float* gain       = (const float*)d_in[3];
    const float* bias       = (const float*)d_in[4];
    const float* ww         = (const float*)d_in[5];
    const float* disp       = (const float*)d_in[6];
    float* out = (float*)d_out;

    const int total_cols = B_DIM * W_DIM * M_DIM;   // 131072
    dim3 grid(total_cols / NCOL);                    // 8192 workgroups
    dim3 block(512);                                 // 16 waves (wave32)
    photonic_net_kernel<<<grid, block, 0, stream>>>(
        field_real, field_imag, conn, gain, bias, ww, disp, out);
}